// ScaledDotProductAttention_13331578487142
// MI455X (gfx1250) — compile-verified
//
#include <hip/hip_runtime.h>
#include <math.h>

// T5-style attention for MI455X (gfx1250, wave32).
// Outputs: [B,H,S,D] attention output, then [B,H,S,S] softmax weights, concat in d_out.
// Memory floor ~ 276 MB stores => ~12us @ 23.3 TB/s; fp32 WMMA keeps compute comparable.

typedef float v2f __attribute__((ext_vector_type(2)));
typedef float v8f __attribute__((ext_vector_type(8)));
typedef unsigned int u32x4 __attribute__((ext_vector_type(4)));
typedef int i32x4 __attribute__((ext_vector_type(4)));
typedef int i32x8 __attribute__((ext_vector_type(8)));

#define NB 2
#define NH 8
#define SEQ 2048
#define DH 64
#define NBUCKETS 32
#define MAXDIST 128
#define RELSPAN (2 * SEQ)   // rel in [-(SEQ-1), SEQ-1] -> index rel + SEQ
#define WPB 4               // waves per block (each wave owns a 16-row Q tile)

// ---------------------------------------------------------------------------
// Kernel 0: precompute bias_rel[h][rel + SEQ] = bias_table[bucket(rel)][h]
// (T5 bidirectional bucketing; 8*4096 floats = 128 KB in d_ws, L2-resident)
// ---------------------------------------------------------------------------
__global__ void t5_bias_precompute(const float* __restrict__ bias_table,
                                   float* __restrict__ bias_rel) {
  int i = blockIdx.x * blockDim.x + threadIdx.x;
  if (i >= NH * RELSPAN) return;
  int h   = i / RELSPAN;
  int rel = (i % RELSPAN) - SEQ;

  int nb = NBUCKETS / 2;              // 16 (bidirectional)
  int rb = (rel > 0) ? nb : 0;
  int rp = rel < 0 ? -rel : rel;
  int max_exact = nb / 2;             // 8
  int bucket;
  if (rp < max_exact) {
    bucket = rp;
  } else {
    float rpf = (float)(rp < 1 ? 1 : rp);
    int large = max_exact +
        (int)(logf(rpf / (float)max_exact) *
              (1.0f / logf((float)MAXDIST / (float)max_exact)) *
              (float)(nb - max_exact));
    bucket = large < (nb - 1) ? large : (nb - 1);
  }
  bias_rel[i] = bias_table[(rb + bucket) * NH + h];
}

// ---------------------------------------------------------------------------
// TDM 1-D contiguous copy: global -> LDS, ndwords 32-bit elements.
// D# built per CDNA5 ISA:
//   group0: [1:0]=count=1, [63:32]=lds_addr, [120:64]=global_addr, [127:126]=type=2
//   group1: data_size=4B, tensor_dim0=tile_dim0=ndwords, tensor_dim1=tile_dim1=1,
//           tensor_dim0_stride=ndwords (1 row). Groups 2/3 zero (<=2D tensor).
// Tracked with TENSORcnt.
// ---------------------------------------------------------------------------
static __device__ __forceinline__ void tdm_load_1d(const void* gsrc,
                                                   unsigned lds_byte_off,
                                                   unsigned ndwords) {
  unsigned long long ga = (unsigned long long)(size_t)gsrc;
  u32x4 g0;
  g0.x = 1u;                                   // count=1 (valid user descriptor)
  g0.y = lds_byte_off;                         // lds_addr (bytes)
  g0.z = (unsigned)ga;                         // global_addr[31:0]
  g0.w = (unsigned)((ga >> 32) & 0x01FFFFFFull) | (2u << 30);  // addr[56:32], type=2
  i32x8 g1;
  g1[0] = 0x00020000;                          // workgroup_mask=0, data_size=2 (4B)
  g1[1] = (int)((ndwords & 0xFFFFu) << 16);    // tensor_dim0[15:0]
  g1[2] = (int)(((ndwords >> 16) & 0xFFFFu) | (1u << 16)); // td0[31:16], tensor_dim1=1
  g1[3] = (int)((ndwords & 0xFFFFu) << 16);    // tile_dim0 = ndwords
  g1[4] = 1;                                   // tile_dim1=1, tile_dim2=0
  g1[5] = (int)ndwords;                        // tensor_dim0_stride[31:0]
  g1[6] = 0;                                   // td0_stride[47:32]=0, td1_stride lo=0
  g1[7] = 0;
  i32x4 gz = {0, 0, 0, 0};
#if __has_include(<hip/amd_detail/amd_gfx1250_TDM.h>)
  // amdgpu-toolchain (clang-23): 6-arg form
  i32x8 gz8 = {0, 0, 0, 0, 0, 0, 0, 0};
  __builtin_amdgcn_tensor_load_to_lds(g0, g1, gz, gz, gz8, 0);
#else
  // ROCm 7.2 (clang-22): 5-arg form
  __builtin_amdgcn_tensor_load_to_lds(g0, g1, gz, gz, 0);
#endif
}

// ---------------------------------------------------------------------------
// Main attention kernel. One wave handles a 16-row Q tile for one (b,h).
// Pass 1: per-lane online softmax stats over all K via f32 WMMA, then a single
//         cross-lane (m,l) merge (log-sum-exp combine) at the end.
// Pass 2: recompute logits, write normalized weights, accumulate O = P @ V.
//
// WMMA f32 16x16x4 layouts (wave32):
//   A 16x4 : lane L holds M = L%16, K = {0,1} + 2*(L/16)  (2 VGPRs, float2)
//   B 4x16 : lane L holds N = L%16, K = {0,1} + 2*(L/16)  (mirror of A)
//   C/D    : vgpr r, lane L -> row M = r + 8*(L/16), col N = L%16
// ---------------------------------------------------------------------------
__global__ __launch_bounds__(WPB * 32)
void t5_attention_kernel(const float* __restrict__ Q, const float* __restrict__ K,
                         const float* __restrict__ V, const int* __restrict__ mask,
                         const float* __restrict__ bias_rel,
                         float* __restrict__ out, float* __restrict__ wts) {
  __shared__ float bias_lds[RELSPAN];    // this head's bias row (16 KB), via TDM
  __shared__ int   mask_lds[SEQ];        // this batch's mask row (8 KB), via TDM
  __shared__ float plds[WPB][16 * 17];   // padded P tile per wave (bank-conflict free)

  const int lane = threadIdx.x & 31;
  const int wave = threadIdx.x >> 5;
  const int b = blockIdx.z, h = blockIdx.y;
  const int q0 = (blockIdx.x * WPB + wave) * 16;

  const size_t bh = (size_t)b * NH + h;
  const float* Qb = Q + bh * SEQ * DH;
  const float* Kb = K + bh * SEQ * DH;
  const float* Vb = V + bh * SEQ * DH;
  float* Wb = wts + (bh * SEQ + (size_t)q0) * SEQ;
  float* Ob = out + (bh * SEQ + (size_t)q0) * DH;

  // Stage bias row + mask row into LDS with the Tensor Data Mover (wave 0 only;
  // wave-uniform scalar branch, descriptors are SGPR-resident).
  if (wave == 0) {
    tdm_load_1d(bias_rel + (size_t)h * RELSPAN,
                (unsigned)(size_t)(void*)bias_lds, RELSPAN);
    tdm_load_1d(mask + (size_t)b * SEQ,
                (unsigned)(size_t)(void*)mask_lds, SEQ);
    __builtin_amdgcn_s_wait_tensorcnt(0);
  }
  __syncthreads();

  const int lo = lane & 15;   // tile column index (and A/B row selector)
  const int hi = lane >> 4;   // half-wave selector

  // Lane's Q row pointer for A-matrix loads (M = lo, K-pair at +2*hi)
  const float* qrow = Qb + (size_t)(q0 + lo) * DH + 2 * hi;

  // Per-lane online softmax state: pm/pe are this lane's running max / sum of
  // exp over the columns it owns, per row r (row M = r + 8*hi).
  float pm[8], pe[8];
#pragma unroll
  for (int r = 0; r < 8; ++r) { pm[r] = -3.0e38f; pe[r] = 0.0f; }
  const float scale = 0.125f;  // 1/sqrt(64)

  // ----------------------------- pass 1: softmax stats ---------------------
  for (int kt = 0; kt < SEQ; kt += 16) {
    const float* krow = Kb + (size_t)(kt + lo) * DH + 2 * hi;
    v8f c = {};
#pragma unroll
    for (int kb = 0; kb < DH; kb += 4) {
      v2f a  = *(const v2f*)(qrow + kb);
      v2f bm = *(const v2f*)(krow + kb);
      c = __builtin_amdgcn_wmma_f32_16x16x4_f32(false, a, false, bm, (short)0, c,
                                                false, false);
    }
    const float madd = -1.0e9f * (float)mask_lds[kt + lo];
    const int relbase = (kt + lo) - q0 - 8 * hi + SEQ;   // rel index for r=0
#pragma unroll
    for (int r = 0; r < 8; ++r) {
      float s = c[r] * scale + bias_lds[relbase - r] + madd;
      float nm = fmaxf(pm[r], s);
      pe[r] = pe[r] * __expf(pm[r] - nm) + __expf(s - nm);
      pm[r] = nm;
    }
  }

  // Cross-lane merge of (max, sum) pairs across the 16 columns of each half:
  // one butterfly total instead of one per K-tile.
#pragma unroll
  for (int r = 0; r < 8; ++r) {
#pragma unroll
    for (int st = 1; st <= 8; st <<= 1) {
      float om = __shfl_xor(pm[r], st, 16);
      float oe = __shfl_xor(pe[r], st, 16);
      float nm = fmaxf(pm[r], om);
      pe[r] = pe[r] * __expf(pm[r] - nm) + oe * __expf(om - nm);
      pm[r] = nm;
    }
  }
  float inv_l[8];
#pragma unroll
  for (int r = 0; r < 8; ++r) inv_l[r] = 1.0f / pe[r];

  v8f acc0 = {}, acc1 = {}, acc2 = {}, acc3 = {};   // O tile, d-chunks 0..3
  float* pl = &plds[wave][0];

  // --------------------- pass 2: weights out + O = P @ V -------------------
  for (int kt = 0; kt < SEQ; kt += 16) {
    const float* krow = Kb + (size_t)(kt + lo) * DH + 2 * hi;
    v8f c = {};
#pragma unroll
    for (int kb = 0; kb < DH; kb += 4) {
      v2f a  = *(const v2f*)(qrow + kb);
      v2f bm = *(const v2f*)(krow + kb);
      c = __builtin_amdgcn_wmma_f32_16x16x4_f32(false, a, false, bm, (short)0, c,
                                                false, false);
    }
    const float madd = -1.0e9f * (float)mask_lds[kt + lo];
    const int relbase = (kt + lo) - q0 - 8 * hi + SEQ;
    float p_[8];
#pragma unroll
    for (int r = 0; r < 8; ++r) {
      float s = c[r] * scale + bias_lds[relbase - r] + madd;
      p_[r] = __expf(s - pm[r]) * inv_l[r];
    }
    // stream normalized weights to global; stage P tile in LDS (row stride 17)
    float* wr = Wb + kt + lo;
#pragma unroll
    for (int r = 0; r < 8; ++r) {
      wr[(size_t)(r + 8 * hi) * SEQ] = p_[r];
      pl[(r + 8 * hi) * 17 + lo] = p_[r];
    }
    // O += P(16x16) @ V(16x64): A = P from LDS, B = V columns from global
    const float* vcol = Vb + (size_t)(kt + 2 * hi) * DH + lo;
#pragma unroll
    for (int kb = 0; kb < 16; kb += 4) {
      v2f a;
      a.x = pl[lo * 17 + kb + 2 * hi + 0];
      a.y = pl[lo * 17 + kb + 2 * hi + 1];
      const float* vp = vcol + (size_t)kb * DH;
      v2f b0; b0.x = vp[0];       b0.y = vp[DH];
      acc0 = __builtin_amdgcn_wmma_f32_16x16x4_f32(false, a, false, b0, (short)0,
                                                   acc0, false, false);
      v2f b1; b1.x = vp[16];      b1.y = vp[DH + 16];
      acc1 = __builtin_amdgcn_wmma_f32_16x16x4_f32(false, a, false, b1, (short)0,
                                                   acc1, false, false);
      v2f b2; b2.x = vp[32];      b2.y = vp[DH + 32];
      acc2 = __builtin_amdgcn_wmma_f32_16x16x4_f32(false, a, false, b2, (short)0,
                                                   acc2, false, false);
      v2f b3; b3.x = vp[48];      b3.y = vp[DH + 48];
      acc3 = __builtin_amdgcn_wmma_f32_16x16x4_f32(false, a, false, b3, (short)0,
                                                   acc3, false, false);
    }
  }

  // write O tile
#pragma unroll
  for (int r = 0; r < 8; ++r) {
    float* orow = Ob + (size_t)(r + 8 * hi) * DH + lo;
    orow[0]  = acc0[r];
    orow[16] = acc1[r];
    orow[32] = acc2[r];
    orow[48] = acc3[r];
  }
}

// ---------------------------------------------------------------------------
extern "C" void kernel_launch(void* const* d_in, const int* in_sizes, int n_in,
                              void* d_out, int out_size, void* d_ws, size_t ws_size,
                              hipStream_t stream) {
  const float* q          = (const float*)d_in[0];  // [B,H,S,D]
  const float* k          = (const float*)d_in[1];  // [B,H,S,D]
  const float* v          = (const float*)d_in[2];  // [B,H,S,D]
  const int*   mask       = (const int*)d_in[3];    // [B,1,1,S]
  const float* bias_table = (const float*)d_in[4];  // [32,H]

  float* out = (float*)d_out;                                   // [B,H,S,D]
  float* wts = out + (size_t)NB * NH * SEQ * DH;                // [B,H,S,S]
  float* bias_rel = (float*)d_ws;                               // [H][2*SEQ] = 128 KB

  t5_bias_precompute<<<(NH * RELSPAN + 255) / 256, 256, 0, stream>>>(bias_table,
                                                                     bias_rel);

  dim3 grid(SEQ / (WPB * 16), NH, NB);   // 32 x 8 x 2
  dim3 block(WPB * 32);                  // 128 threads = 4 waves
  t5_attention_kernel<<<grid, block, 0, stream>>>(q, k, v, mask, bias_rel, out, wts);
}